// ChamferLoss_77094662964081
// MI455X (gfx1250) — compile-verified
//
#include <hip/hip_runtime.h>

typedef float v2f __attribute__((ext_vector_type(2)));
typedef float v8f __attribute__((ext_vector_type(8)));

#define WAVES_PER_BLOCK 8
#define TILE 16

// One wave owns a 16-row strip of `src` (one batch) and computes, for each of
// those 16 rows, min over all M `dst` points of clamp(||s-d||^2, 0).
// Cross terms come from V_WMMA_F32_16X16X4_F32 (K=3 padded to 4 with zeros).
__global__ void __launch_bounds__(WAVES_PER_BLOCK * 32)
chamfer_rowmin_kernel(const float* __restrict__ src,
                      const float* __restrict__ dst,
                      float* __restrict__ rowmin,
                      int N, int M, int numWaves)
{
    const int lane = threadIdx.x & 31;
    const int half = lane >> 4;    // 0: lanes 0-15, 1: lanes 16-31
    const int idx  = lane & 15;
    const int wave = blockIdx.x * (blockDim.x >> 5) + (threadIdx.x >> 5);
    if (wave >= numWaves) return;  // wave-uniform: EXEC stays all-1s for WMMA

    const int tilesPerBatch = N >> 4;
    const int b       = wave / tilesPerBatch;
    const int rowbase = (wave % tilesPerBatch) << 4;

    // ---- A fragment: 16x4 fp32. Lane holds src row (rowbase+idx);
    //      half 0 supplies K=0,1 = (x,y); half 1 supplies K=2,3 = (z,0).
    const float* sp = src + ((size_t)b * N + rowbase + idx) * 3;
    float sx = sp[0], sy = sp[1], sz = sp[2];
    v2f a;
    a.x = half ? sz : sx;
    a.y = half ? 0.0f : sy;
    float s2_own = sx * sx + sy * sy + sz * sz;

    // s2 for the 8 rows mapped to this lane's C/D VGPRs (M = 8*half + r).
    float s2arr[8];
    #pragma unroll
    for (int r = 0; r < 8; ++r)
        s2arr[r] = __shfl(s2_own, half * 8 + r, 32);

    float mn[8];
    #pragma unroll
    for (int r = 0; r < 8; ++r) mn[r] = 3.402823466e38f;

    const float* dbase = dst + (size_t)b * M * 3;
    for (int colbase = 0; colbase < M; colbase += TILE) {
        // ---- B fragment: 4x16 fp32, column = dst point (colbase+idx).
        const float* dp = dbase + (size_t)(colbase + idx) * 3;
        float dx = dp[0], dy = dp[1], dz = dp[2];
        v2f bf;
        bf.x = half ? dz : dx;
        bf.y = half ? 0.0f : dy;
        float d2 = dx * dx + dy * dy + dz * dz;

        v8f c = {};
        c = __builtin_amdgcn_wmma_f32_16x16x4_f32(
                /*neg_a=*/false, a, /*neg_b=*/false, bf,
                /*c_mod=*/(short)0, c, /*reuse_a=*/false, /*reuse_b=*/false);

        // dis[m][n] = s2[m] + d2[n] - 2*cross; clamp >= 0; running row-min.
        #pragma unroll
        for (int r = 0; r < 8; ++r) {
            float t   = s2arr[r] + d2;
            float dis = __builtin_fmaf(-2.0f, c[r], t);
            dis = fmaxf(dis, 0.0f);
            mn[r] = fminf(mn[r], dis);
        }
    }

    // Min across the 16 lanes of each half (each lane covered cols == idx mod 16).
    #pragma unroll
    for (int mask = 1; mask < 16; mask <<= 1) {
        #pragma unroll
        for (int r = 0; r < 8; ++r)
            mn[r] = fminf(mn[r], __shfl_xor(mn[r], mask, 32));
    }

    if (idx == 0) {
        float* outp = rowmin + (size_t)b * N + rowbase + half * 8;
        #pragma unroll
        for (int r = 0; r < 8; ++r) outp[r] = mn[r];
    }
}

__global__ void __launch_bounds__(256)
chamfer_reduce_kernel(const float* __restrict__ rowA,
                      const float* __restrict__ rowB,
                      float* __restrict__ out,
                      int lenA, int lenB, float scaleA, float scaleB)
{
    __shared__ float sm[256];
    float acc = 0.0f;
    for (int i = threadIdx.x; i < lenA; i += blockDim.x) acc += rowA[i] * scaleA;
    for (int i = threadIdx.x; i < lenB; i += blockDim.x) acc += rowB[i] * scaleB;
    sm[threadIdx.x] = acc;
    __syncthreads();
    for (int s = 128; s > 0; s >>= 1) {
        if ((int)threadIdx.x < s) sm[threadIdx.x] += sm[threadIdx.x + s];
        __syncthreads();
    }
    if (threadIdx.x == 0) out[0] = sm[0];
}

extern "C" void kernel_launch(void* const* d_in, const int* in_sizes, int n_in,
                              void* d_out, int out_size, void* d_ws, size_t ws_size,
                              hipStream_t stream) {
    const float* f  = (const float*)d_in[0];
    const float* f_ = (const float*)d_in[1];
    float* out = (float*)d_out;

    const int B = 4;
    const int N = in_sizes[0] / (B * 3);   // 8192
    const int M = in_sizes[1] / (B * 3);   // 8192

    float* rowA = (float*)d_ws;            // [B*N] min over f_ for each f point
    float* rowB = rowA + (size_t)B * N;    // [B*M] min over f for each f_ point

    const int wavesA  = B * (N / 16);
    const int blocksA = (wavesA + WAVES_PER_BLOCK - 1) / WAVES_PER_BLOCK;
    chamfer_rowmin_kernel<<<blocksA, WAVES_PER_BLOCK * 32, 0, stream>>>(
        f, f_, rowA, N, M, wavesA);

    const int wavesB  = B * (M / 16);
    const int blocksB = (wavesB + WAVES_PER_BLOCK - 1) / WAVES_PER_BLOCK;
    chamfer_rowmin_kernel<<<blocksB, WAVES_PER_BLOCK * 32, 0, stream>>>(
        f_, f, rowB, M, N, wavesB);

    chamfer_reduce_kernel<<<1, 256, 0, stream>>>(
        rowA, rowB, out, B * N, B * M, 1.0f / (B * N), 1.0f / (B * M));
}